// VocabParallelEmbeddingWithLoRA_893353198244
// MI455X (gfx1250) — compile-verified
//
#include <hip/hip_runtime.h>

#define ORG_VOCAB   32000
#define EXTRA_VOCAB 256
#define VPL         32256   // ORG_VOCAB + EXTRA_VOCAB
#define EMBED_DIM   4096
#define RANK        16
#define MAX_LORAS   8

typedef __attribute__((ext_vector_type(2))) float v2f;
typedef __attribute__((ext_vector_type(8))) float v8f;

// out[t, d] = base_weight[x_mod[t], d] + sum_{l,r} A'[t,(l,r)] * B'[(l,r), d]
//   A'[t,(l,r)] = (lora_idx[t]==l) ? lora_a[x[t]+l*VPL, r] : 0     (16 x 128 per token block)
//   B'[(l,r),d] = lora_b[l, 0, d, r]                               (128 x 4096, L2-resident)
// One wave = one 16-wide column tile; B' fragments held in 64 VGPRs across token blocks.
// Output is write-only -> non-temporal stores keep L2 free for the gather tables.
__global__ __launch_bounds__(256) void lora_embed_wmma_kernel(
    const int*   __restrict__ x,
    const int*   __restrict__ lidx,
    const float* __restrict__ base_w,
    const float* __restrict__ lora_a,
    const float* __restrict__ lora_b,
    float*       __restrict__ out,
    int ntb, int nsplit)
{
    const int lane = threadIdx.x & 31;
    const int wib  = threadIdx.x >> 5;
    const int gw   = blockIdx.x * 8 + wib;   // global wave id
    const int ct   = gw & 255;               // column tile: d0 = ct*16 (256 tiles cover D=4096)
    const int ts   = gw >> 8;                // token-block split id
    const int d0   = ct << 4;
    const int n    = lane & 15;              // N / column within tile
    const int half = lane >> 4;              // wave half (K sub-slot for A/B frags)

    // ---- B' fragments for this column tile, K = 0..127 (held in registers) ----
    // bfrag[2*kc+jj] holds B'[4*kc + 2*half + jj][d0+n]  (ISA B 4x16 f32 layout)
    float bfrag[64];
#pragma unroll
    for (int kc = 0; kc < 32; ++kc) {
#pragma unroll
        for (int jj = 0; jj < 2; ++jj) {
            const int k = 4 * kc + 2 * half + jj;
            const int l = k >> 4;
            const int r = k & 15;
            bfrag[2 * kc + jj] =
                lora_b[(size_t)l * (EMBED_DIM * RANK) + (size_t)(d0 + n) * RANK + r];
        }
    }

    for (int tb = ts; tb < ntb; tb += nsplit) {
        // Lane owns token (lane & 15) of this block for the A fragment.
        const int t  = (tb << 4) + n;
        const int xt = x[t];
        const int li = lidx[t];

        // A-row values this lane will feed into A slots:
        // areg[2*p+jj] = lora_a_row[4*p + 2*half + jj]  (compile-time indexed at use)
        const float* arow =
            lora_a + ((size_t)xt + (size_t)li * VPL) * RANK + 2 * half;
        float areg[8];
#pragma unroll
        for (int p = 0; p < 4; ++p) {
            areg[2 * p + 0] = arow[4 * p + 0];
            areg[2 * p + 1] = arow[4 * p + 1];
        }

        // ---- base-embedding gather straight into the C accumulator ----
        // C layout: VGPR i -> row M = 8*half + i, column N = n
        v8f c;
#pragma unroll
        for (int i = 0; i < 8; ++i) {
            const int src = (half << 3) + i;        // token-in-block for this C row
            const int xb  = __shfl(xt, src, 32);
            const int lb  = __shfl(li, src, 32);
            const int xm  = xb + ((xb >= ORG_VOCAB) ? lb * EXTRA_VOCAB : 0);
            c[i] = base_w[(size_t)xm * EMBED_DIM + d0 + n];
        }

        // ---- K=128 accumulation: 32 x v_wmma_f32_16x16x4_f32 ----
#pragma unroll
        for (int kc = 0; kc < 32; ++kc) {
            const int l = kc >> 2;   // compile-time lora id for this K chunk
            const int p = kc & 3;    // compile-time rank sub-block
            const bool eq = (li == l);
            v2f a, b;
            a.x = eq ? areg[2 * p + 0] : 0.0f;
            a.y = eq ? areg[2 * p + 1] : 0.0f;
            b.x = bfrag[2 * kc + 0];
            b.y = bfrag[2 * kc + 1];
            c = __builtin_amdgcn_wmma_f32_16x16x4_f32(
                    false, a, false, b, (short)0, c, false, false);
        }

        // ---- store 16x16 output tile (non-temporal: write-only stream) ----
        float* orow = out + (size_t)((tb << 4) + (half << 3)) * EMBED_DIM + d0 + n;
#pragma unroll
        for (int i = 0; i < 8; ++i) {
            __builtin_nontemporal_store(c[i], orow + (size_t)i * EMBED_DIM);
        }
    }
}

extern "C" void kernel_launch(void* const* d_in, const int* in_sizes, int n_in,
                              void* d_out, int out_size, void* d_ws, size_t ws_size,
                              hipStream_t stream) {
    const int*   x  = (const int*)  d_in[0];   // [T]
    const int*   li = (const int*)  d_in[1];   // [T]
    const float* bw = (const float*)d_in[2];   // [VPL, D]
    const float* la = (const float*)d_in[3];   // [MAX_LORAS*VPL, RANK]
    const float* lb = (const float*)d_in[4];   // [MAX_LORAS, 1, D, RANK]
    float* out = (float*)d_out;

    const int T   = in_sizes[0];
    const int ntb = T / 16;                    // token blocks (T multiple of 16)
    const int S   = (ntb < 64) ? ntb : 64;     // token-split factor
    const int waves  = 256 * S;                // 256 column tiles x S splits
    const int blocks = waves / 8;              // 8 waves (256 threads) per block

    lora_embed_wmma_kernel<<<blocks, 256, 0, stream>>>(x, li, bw, la, lb, out, ntb, S);
}